// ConvCaps_9337258901599
// MI455X (gfx1250) — compile-verified
//
#include <hip/hip_runtime.h>
#include <math.h>

// ---------------------------------------------------------------------------
// ConvCaps routing-by-agreement, fused, MI455X / gfx1250.
//
//  - grid.x = 57 workgroups, one 16-row M-tile of the b*l = 900 rows each
//    (buffers padded to 912 rows so hot loops need no bounds checks).
//  - 1024 threads = 32 wave32 waves; wave w owns output capsule B = w.
//  - pose patches in 288 KB LDS: pA[k'(288)][m(16)][c(16)], loaded once.
//  - votes recomputed per pass: 4 chained V_WMMA_F32_16X16X4_F32 (K = C = 16);
//    s / v / vote tiles live in WMMA C-layout VGPRs the whole time.
//  - squash + logit dots: 16-lane __shfl_xor butterflies over the D dim.
//  - routing state in workspace, layout [wg][k'][B][m_local] so the c-read is
//    one coalesced 64B load + shfl broadcast, and the logit RMW is one
//    coalesced 16-lane access.  Both buffers are L2-resident (2 x 33.6 MB).
// ---------------------------------------------------------------------------

typedef float v2f __attribute__((ext_vector_type(2)));
typedef float v8f __attribute__((ext_vector_type(8)));

#define AIN   32
#define BOUT  32
#define CSZ   16
#define KKA   288          // 9 * 32
#define OHW   15
#define LSZ   225          // 15*15
#define MTOT  900          // 4 * 225
#define NCOL  512          // BOUT * 16
#define MTILES 57
#define CAPS_EPS 1e-8f

// routing buffer: [MTILES][KKA][BOUT][16]  (padded rows are harmless garbage)
#define RBUF_ELEMS ((size_t)MTILES * KKA * BOUT * 16)   // 8,404,992 floats

__device__ __forceinline__ float halfsum16(float x) {
  // sum across the 16 lanes of a half-wave (the D dimension of a C-layout tile)
  x += __shfl_xor(x, 1, 32);
  x += __shfl_xor(x, 2, 32);
  x += __shfl_xor(x, 4, 32);
  x += __shfl_xor(x, 8, 32);
  return x;
}

__global__ __launch_bounds__(1024)
void convcaps_routing_kernel(const float* __restrict__ pose,   // (4, 512, 32, 32)
                             const float* __restrict__ Wt,     // (288, 512, 16)
                             float* __restrict__ out,          // (4, 512, 15, 15)
                             float* __restrict__ logits,       // ws rbuf
                             float* __restrict__ cbuf)         // ws rbuf
{
  extern __shared__ float pA[];                 // [KKA][16][16] = 288 KB
  const int tid  = threadIdx.x;
  const int lane = tid & 31;
  const int wv   = tid >> 5;                    // wave id == output capsule B
  const int hi   = lane >> 4;                   // half-wave select (M +8)
  const int m0   = blockIdx.x * 16;
  const size_t wgbase = (size_t)blockIdx.x * KKA * BOUT * 16;

  // ---- gather pose patches into LDS: pA[k'][m][c] --------------------------
  for (int idx = tid; idx < KKA * 256; idx += 1024) {
    const int c  = idx & 15;
    const int mm = (idx >> 4) & 15;
    const int kp = idx >> 8;
    const int kk = kp / AIN;
    const int a  = kp % AIN;
    const int ki = kk / 3, kj = kk % 3;
    int m = m0 + mm; if (m >= MTOT) m = MTOT - 1;      // clamp: padded rows dup row 899
    const int bi = m / LSZ;
    const int l  = m % LSZ;
    const int oy = l / OHW, ox = l % OHW;
    pA[idx] = pose[(((size_t)bi * 512 + a * 16 + c) * 32 + (oy * 2 + ki)) * 32
                   + (ox * 2 + kj)];
  }
  __syncthreads();

  // votes tile for (k', capsule wv): D[m,n] = sum_c pA[k'][m][c] * W[k'][n][c]
  auto votes_tile = [&](int kp) -> v8f {
    v8f acc = {0.f, 0.f, 0.f, 0.f, 0.f, 0.f, 0.f, 0.f};
    const float* arow = &pA[(kp * 16 + (lane & 15)) * 16];
    const float* brow = &Wt[((size_t)kp * NCOL + wv * 16 + (lane & 15)) * CSZ];
    if (kp + 1 < KKA)
      __builtin_prefetch(&Wt[((size_t)(kp + 1) * NCOL + wv * 16) * CSZ], 0, 3);
#pragma unroll
    for (int t = 0; t < 4; ++t) {
      const int ck = t * 4 + hi * 2;
      v2f a = *(const v2f*)(arow + ck);
      v2f b = *(const v2f*)(brow + ck);
      acc = __builtin_amdgcn_wmma_f32_16x16x4_f32(false, a, false, b,
                                                  (short)0, acc, false, false);
    }
    return acc;
  };

  // ---- pass0: softmax over B of raw logits -> cbuf ------------------------
  auto softmax_pass = [&]() {
    for (int r = tid; r < 16 * KKA; r += 1024) {
      const int j  = r & 15;                    // m_local (coalesced over j)
      const int kp = r >> 4;
      const size_t base = wgbase + (size_t)kp * BOUT * 16 + j;
      float vals[BOUT];
      float mx = -1e30f;
#pragma unroll
      for (int b = 0; b < BOUT; ++b) { vals[b] = logits[base + b * 16]; mx = fmaxf(mx, vals[b]); }
      float sum = 0.f;
#pragma unroll
      for (int b = 0; b < BOUT; ++b) { vals[b] = __expf(vals[b] - mx); sum += vals[b]; }
      const float inv = 1.f / sum;
#pragma unroll
      for (int b = 0; b < BOUT; ++b) cbuf[base + b * 16] = vals[b] * inv;
    }
    __threadfence();
    __syncthreads();
  };

  // ---- pass1: s[m,d] = sum_k' c[m,k',B] * votes[m,k',B,d] -----------------
  auto accum_pass = [&](bool uniform) -> v8f {
    v8f s = {0.f, 0.f, 0.f, 0.f, 0.f, 0.f, 0.f, 0.f};
    for (int kp = 0; kp < KKA; ++kp) {
      v8f vt = votes_tile(kp);
      float creg = 0.f;
      if (!uniform)                             // one coalesced 64B read / (k',B)
        creg = cbuf[wgbase + ((size_t)kp * BOUT + wv) * 16 + (lane & 15)];
#pragma unroll
      for (int j = 0; j < 8; ++j) {
        const float cv = uniform ? (1.0f / 32.0f)          // softmax of zeros
                                 : __shfl(creg, j + hi * 8, 32);
        s[j] += cv * vt[j];
      }
    }
    return s;
  };

  // ---- squash: v = s * |s|^2 / ((1+|s|^2) * sqrt(|s|^2 + eps)) ------------
  auto squash = [&](const v8f& s) -> v8f {
    v8f v = {0.f, 0.f, 0.f, 0.f, 0.f, 0.f, 0.f, 0.f};
#pragma unroll
    for (int j = 0; j < 8; ++j) {
      const float n2 = halfsum16(s[j] * s[j]);
      const float sc = n2 / ((1.f + n2) * sqrtf(n2 + CAPS_EPS));
      v[j] = s[j] * sc;
    }
    return v;
  };

  // ---- pass2: logits[m,k',B] += sum_d v[m,B,d]*votes[m,k',B,d] ------------
  auto update_pass = [&](const v8f& vfrag, bool first) {
    for (int kp = 0; kp < KKA; ++kp) {
      v8f vt = votes_tile(kp);
      float xs[8];
#pragma unroll
      for (int j = 0; j < 8; ++j) xs[j] = halfsum16(vfrag[j] * vt[j]);
      // transpose (register j, half hi) -> lane-major m_local = lane for a
      // single coalesced 16-lane RMW: lane L needs xs[L&7] from half (L>>3).
      const int src = ((lane >> 3) & 1) * 16;
#pragma unroll
      for (int j = 0; j < 8; ++j) xs[j] = __shfl(xs[j], src, 32);
      const int ls = lane & 7;
      float val = xs[0];
#pragma unroll
      for (int j = 1; j < 8; ++j) val = (ls == j) ? xs[j] : val;
      if (lane < 16) {
        const size_t a = wgbase + ((size_t)kp * BOUT + wv) * 16 + lane;
        logits[a] = first ? val : (logits[a] + val);       // no init needed
      }
    }
    __threadfence();
    __syncthreads();
  };

  // ---- 3 routing iterations, manually peeled ------------------------------
  v8f s, vfrag;
  // iter 0: c is uniform 1/32 (softmax of zero logits); logits '=' write
  s = accum_pass(true);
  vfrag = squash(s);
  update_pass(vfrag, true);
  // iter 1
  softmax_pass();
  s = accum_pass(false);
  vfrag = squash(s);
  update_pass(vfrag, false);
  // iter 2 (final: no logit update)
  softmax_pass();
  s = accum_pass(false);
  vfrag = squash(s);

  // ---- store v: out[bi][B*16 + d][oy][ox] ---------------------------------
#pragma unroll
  for (int j = 0; j < 8; ++j) {
    const int m = m0 + j + hi * 8;
    if (m < MTOT) {
      const int bi = m / LSZ;
      const int l  = m % LSZ;
      const int n  = wv * 16 + (lane & 15);
      out[((size_t)bi * NCOL + n) * LSZ + l] = vfrag[j];
    }
  }
}

extern "C" void kernel_launch(void* const* d_in, const int* in_sizes, int n_in,
                              void* d_out, int out_size, void* d_ws, size_t ws_size,
                              hipStream_t stream) {
  const float* pose = (const float*)d_in[0];   // (4, 512, 32, 32) f32
  const float* Wt   = (const float*)d_in[1];   // (288, 512, 16)  f32
  float* out = (float*)d_out;                  // (4, 512, 15, 15) f32

  float* logits = (float*)d_ws;                // 33.6 MB
  float* cbuf   = logits + RBUF_ELEMS;         // 33.6 MB (need ws >= 67.3 MB)

  const size_t lds_bytes = (size_t)KKA * 16 * 16 * sizeof(float);  // 288 KB
  convcaps_routing_kernel<<<MTILES, 1024, lds_bytes, stream>>>(
      pose, Wt, out, logits, cbuf);
}